// MCPBRNN_PETconstraint_MassRelax_Regular_MI_outputloss_sigmoid_53695681134745
// MI455X (gfx1250) — compile-verified
//
#include <hip/hip_runtime.h>
#include <stdint.h>
#include <math.h>

// ---------------- constants from the reference ----------------
#define SCALE_MR_C 500.0f
#define ML_C 2.9086f
#define SL_C 1.898f
#define SPIN_LEN_C 365
#define TRAIN_LEN_C 6000
#define NB_MAX 8192

typedef unsigned int v4u __attribute__((ext_vector_type(4)));
typedef int          v8i __attribute__((ext_vector_type(8)));
typedef int          v4i __attribute__((ext_vector_type(4)));

// Uniform precomputed parameters (affine forms of the sigmoid/tanh args)
struct Params {
    float ao, al;          // e_o/denom, e_l/denom
    float k1, b_oo;        // oo sigmoid arg = c*k1 + b_oo   (c0p1 term folded in)
    float k2, k3, b_ol;    // ol sigmoid arg = c*k2 + u2*k3 + b_ol
    float sv;              // sigmoid(weight_r_yvm)
    float eb;              // exp(bias_b0_yrm)
    float es;              // exp(weight_s_yvm)
    float thresh;          // eb * 500
};

struct Gates { float oo, ol, olc, f, ov, mr, c1; };

__device__ __forceinline__ float sigmoidf_(float x) {
    return 1.0f / (1.0f + expf(-x));
}

__device__ __forceinline__ Params make_params(
    float mo, float so, float wro, float wrl, float wrf, float wrv,
    float b0o, float w1o, float w2o, float b0l, float w1l, float w2l,
    float wsv, float b0r)
{
    Params p;
    float eo = expf(wro), el = expf(wrl), ef = expf(wrf);
    float inv_d = 1.0f / (eo + el + ef);
    p.ao = eo * inv_d;
    p.al = el * inv_d;
    float inv_so = 1.0f / so;
    p.k1   = w1o * inv_so;
    // bias + (c-mo)/so*w1 + (0-mo)/so*w2  ->  c*k1 + b_oo
    p.b_oo = b0o - mo * inv_so * w1o - mo * inv_so * w2o;
    p.k2   = w1l * inv_so;
    p.k3   = w2l / SL_C;
    p.b_ol = b0l - mo * inv_so * w1l - (ML_C / SL_C) * w2l;
    p.sv   = sigmoidf_(wrv);
    p.eb   = expf(b0r);
    p.es   = expf(wsv);
    p.thresh = p.eb * SCALE_MR_C;
    return p;
}

__device__ __forceinline__ Gates compute_gates(float c, float u1, float u2, const Params& p) {
    Gates g;
    g.oo  = p.ao * sigmoidf_(fmaf(c, p.k1, p.b_oo));
    g.ol  = p.al * sigmoidf_(fmaf(c, p.k2, fmaf(u2, p.k3, p.b_ol)));
    // olc = c>0 ? ol - relu(ol - u2/c) : ol   ==  c>0 ? min(ol, u2/c) : ol
    g.olc = (c > 0.0f) ? fminf(g.ol, u2 / c) : g.ol;
    g.f   = 1.0f - g.oo - g.olc;
    float ov1 = p.sv * tanhf((c * (1.0f / SCALE_MR_C) - p.eb) * p.es);
    g.ov  = fminf(ov1, g.f);             // ov1 - relu(ov1 - f)
    g.mr  = g.ov * fabsf(c - p.thresh);
    g.c1  = fmaf(g.f, c, u1) - g.mr;
    return g;
}

__device__ __forceinline__ double shfl_down_dbl(double x, int off) {
    union { double d; int i[2]; } u;
    u.d = x;
    u.i[0] = __shfl_down(u.i[0], off, 32);
    u.i[1] = __shfl_down(u.i[1], off, 32);
    return u.d;
}

__global__ void __launch_bounds__(1024, 1)
mcpbrnn_scan_kernel(const float* __restrict__ x, const float* __restrict__ y_obs,
                    const float* __restrict__ p_mean, const float* __restrict__ p_std,
                    const int*   __restrict__ time_lag_p,
                    const float* __restrict__ w_r_yom, const float* __restrict__ w_r_ylm,
                    const float* __restrict__ w_r_yfm, const float* __restrict__ w_r_yvm,
                    const float* __restrict__ b0_yom,  const float* __restrict__ w_b1_yom,
                    const float* __restrict__ w_b2_yom,
                    const float* __restrict__ b0_ylm,  const float* __restrict__ w_b1_ylm,
                    const float* __restrict__ w_b2_ylm,
                    const float* __restrict__ w_s_yvm, const float* __restrict__ b0_yrm,
                    float* __restrict__ out, int nB, int nY)
{
    __shared__ float2 s_x[NB_MAX];   // phase1: (u1,u2); after phase2, .x recycled to hold c_i
    __shared__ float  s_std;

    const int tid = threadIdx.x;
    if (nB > NB_MAX) nB = NB_MAX;

    // ---- Phase 1a: TDM async DMA of x (2*nB f32) into LDS (issued once; EXEC-independent)
    if (tid == 0) {
        uint32_t lds_off = (uint32_t)(uintptr_t)(&s_x[0]);   // low 32 bits of generic LDS ptr == LDS byte offset
        uint64_t ga = (uint64_t)(uintptr_t)x;
        uint32_t n0 = (uint32_t)(2 * nB);                    // elements (fits 16-bit tile_dim0)
        // D# group 0: count=1, is_restore=0, lds_addr, global_addr[56:0], type=2
        v4u g0;
        g0[0] = 1u;
        g0[1] = lds_off;
        g0[2] = (uint32_t)ga;
        g0[3] = ((uint32_t)(ga >> 32) & 0x01FFFFFFu) | 0x80000000u;  // type=2 in bits[127:126]
        // D# group 1: mask=0, data_size=4B(code 2), no barrier/iterate/pad,
        //             tensor_dim0=n0, tensor_dim1=1, tile_dim0=n0, tile_dim1=1, tile_dim2=0,
        //             tensor_dim0_stride=n0, tensor_dim1_stride=n0
        v8i g1;
        g1[0] = 0x00020000;                                   // data_size=2 (4 bytes)
        g1[1] = (int)((n0 & 0xFFFFu) << 16);                  // abar_addr=0 | tensor_dim0[15:0]
        g1[2] = (int)(((n0 >> 16) & 0xFFFFu) | (1u << 16));   // tensor_dim0[31:16] | tensor_dim1=1
        g1[3] = (int)((n0 & 0xFFFFu) << 16);                  // tensor_dim1 hi=0 | tile_dim0
        g1[4] = 1;                                            // tile_dim1=1 | tile_dim2=0
        g1[5] = (int)n0;                                      // tensor_dim0_stride[31:0]
        g1[6] = (int)((n0 & 0xFFFFu) << 16);                  // stride0 hi=0 | stride1[15:0]
        g1[7] = (int)(n0 >> 16);                              // stride1[47:16]
        v4i g2 = {0, 0, 0, 0};
        v4i g3 = {0, 0, 0, 0};
        v8i g4 = {0, 0, 0, 0, 0, 0, 0, 0};                    // extra operand on clang-23 form
        __builtin_amdgcn_tensor_load_to_lds(g0, g1, g2, g3, g4, 0);
    }

    // ---- Phase 1b: redundant manual copy (identical bytes -> benign race with the TDM)
    {
        const float4* xv = (const float4*)x;
        float4* sxv = (float4*)s_x;
        int nv4 = (2 * nB) >> 2;
        for (int i = tid; i < nv4; i += blockDim.x) sxv[i] = xv[i];
    }

    // Uniform scalars / precomputed params (every thread; cheap, all scalar loads)
    const int lag = *time_lag_p;
    const Params p = make_params(*p_mean, *p_std, *w_r_yom, *w_r_ylm, *w_r_yfm, *w_r_yvm,
                                 *b0_yom, *w_b1_yom, *w_b2_yom,
                                 *b0_ylm, *w_b1_ylm, *w_b2_ylm, *w_s_yvm, *b0_yrm);

    if (tid == 0) {
        __builtin_amdgcn_s_wait_tensorcnt(0);   // TDM done before anyone consumes LDS
    }
    __syncthreads();

    // ---- Phase 2 (concurrent):
    //  wave 0 / lane 0: the inherently-serial nonlinear scan, c_i stored back into s_x[i].x
    //  wave 1: deterministic wave32 shuffle reduction for std(y_obs[365:6000], ddof=1)
    if (tid == 0) {
        float c = 0.0f;
        float2 uv = s_x[0];
        for (int i = 0; i < nB; ++i) {
            float2 uv_next = (i + 1 < nB) ? s_x[i + 1] : uv;   // hoisted off the critical chain
            Gates g = compute_gates(c, uv.x, uv.y, p);
            s_x[i].x = c;                                      // state consumed by phase 3
            c = (i >= lag) ? g.c1 : c;
            uv = uv_next;
        }
    } else if (tid >= 32 && tid < 64) {
        int lane = tid - 32;
        int hi = (TRAIN_LEN_C < nY) ? TRAIN_LEN_C : nY;
        __builtin_prefetch(&y_obs[SPIN_LEN_C + lane], 0, 3);
        double s = 0.0, s2 = 0.0;
        for (int i = SPIN_LEN_C + lane; i < hi; i += 32) {
            double v = (double)y_obs[i];
            s += v; s2 += v * v;
        }
        for (int off = 16; off; off >>= 1) {
            s  += shfl_down_dbl(s,  off);
            s2 += shfl_down_dbl(s2, off);
        }
        if (lane == 0) {
            double n = (double)(hi - SPIN_LEN_C);
            s_std = (float)sqrt((s2 - s * s / n) / (n - 1.0));
        }
    }
    __syncthreads();

    // ---- Phase 3: wide, coalesced re-evaluation of gates at each stored c_i + all 14 outputs
    const float obsstd = s_std;
    for (int i = tid; i < nB; i += blockDim.x) {
        float c  = s_x[i].x;          // state from the scan
        float u2 = x[2 * i + 1];      // u1 not needed for outputs
        Gates g = compute_gates(c, /*u1=*/0.0f, u2, p);
        float m  = (i >= lag) ? 1.0f : 0.0f;
        float hn = m * g.oo * c;
        float os = m * obsstd;
        out[ 0 * nB + i]     = hn;           // h_n
        out[ 1 * nB + i]     = m * c;        // c_n
        out[ 2 * nB + i]     = m * g.ol * c; // l_n
        out[ 3 * nB + i]     = m * g.olc * c;// lc_n
        out[ 4 * nB + i]     = 0.0f;         // bp_n
        out[ 5 * nB + i]     = 0.0f;         // Gate_ib
        out[ 6 * nB + i]     = m * g.oo;     // Gate_oo
        out[ 7 * nB + i]     = m * g.ol;     // Gate_ol
        out[ 8 * nB + i]     = m * g.olc;    // Gate_olc
        out[ 9 * nB + i]     = m * g.f;      // Gate_f
        out[10 * nB + 2 * i]     = hn;       // h_nout[:,0]
        out[10 * nB + 2 * i + 1] = os;       // h_nout[:,1]
        out[12 * nB + i]     = os;           // obs_std
        out[13 * nB + i]     = m * g.ov;     // Gate_ov
        out[14 * nB + i]     = m * g.mr;     // mr_n
    }
}

extern "C" void kernel_launch(void* const* d_in, const int* in_sizes, int n_in,
                              void* d_out, int out_size, void* d_ws, size_t ws_size,
                              hipStream_t stream) {
    (void)n_in; (void)out_size; (void)d_ws; (void)ws_size;
    const float* x        = (const float*)d_in[0];
    const float* y_obs    = (const float*)d_in[1];
    const float* p_mean   = (const float*)d_in[2];
    const float* p_std    = (const float*)d_in[3];
    /* d_in[4] = epoch (unused) */
    const int*   time_lag = (const int*)  d_in[5];
    const float* w_r_yom  = (const float*)d_in[6];
    const float* w_r_ylm  = (const float*)d_in[7];
    const float* w_r_yfm  = (const float*)d_in[8];
    const float* w_r_yvm  = (const float*)d_in[9];
    const float* b0_yom   = (const float*)d_in[10];
    const float* w_b1_yom = (const float*)d_in[11];
    const float* w_b2_yom = (const float*)d_in[12];
    const float* b0_ylm   = (const float*)d_in[13];
    const float* w_b1_ylm = (const float*)d_in[14];
    const float* w_b2_ylm = (const float*)d_in[15];
    const float* w_s_yvm  = (const float*)d_in[16];
    const float* b0_yrm   = (const float*)d_in[17];

    int nB = in_sizes[0] / 2;   // x is (B,1,2)
    int nY = in_sizes[1];       // y_obs is (B,1)

    mcpbrnn_scan_kernel<<<1, 1024, 0, stream>>>(
        x, y_obs, p_mean, p_std, time_lag,
        w_r_yom, w_r_ylm, w_r_yfm, w_r_yvm,
        b0_yom, w_b1_yom, w_b2_yom,
        b0_ylm, w_b1_ylm, w_b2_ylm,
        w_s_yvm, b0_yrm,
        (float*)d_out, nB, nY);
}